// IterModel_21775484190807
// MI455X (gfx1250) — compile-verified
//
#include <hip/hip_runtime.h>
#include <hip/hip_bf16.h>
#include <stdint.h>

typedef __attribute__((ext_vector_type(2)))  float    v2f;
typedef __attribute__((ext_vector_type(8)))  float    v8f;

__device__ __forceinline__ float fast_tanh(float x) {
  // tanh(x) = (e-1)/(e+1), e = exp(2x) = exp2(x * 2*log2(e)); clamp avoids inf/inf
  float xc = fminf(fmaxf(x, -10.f), 10.f);
  float e  = __builtin_amdgcn_exp2f(xc * 2.8853900817779268f);
  return (e - 1.0f) * __builtin_amdgcn_rcpf(e + 1.0f);
}

__device__ __forceinline__ void async_b128(uint32_t lds_off, const float* gaddr) {
  // GLOBAL_LOAD_ASYNC_TO_LDS_B128: each lane copies 16 bytes from the global address in vaddr to the LDS address in vdst, ASYNCcnt-tracked
  asm volatile("global_load_async_to_lds_b128 %0, %1, off"
               :: "v"(lds_off), "v"(gaddr) : "memory");
}
__device__ __forceinline__ void wait_async0() {
  asm volatile("s_wait_asynccnt 0x0" ::: "memory");
}

// ---------------------------------------------------------------------------
// LDS-staged GEMM for C[M x 256] = A[M x 256] @ B[256 x 256] (+bias).
// One block (8 waves) per 16-row stripe.  K is chunked by 64: A(16x64) and
// B(64x256) are staged into LDS via global_load_async_to_lds_b128, then each
// wave computes two 16x16 N-tiles (cols w*16 and w*16+128) with
// V_WMMA_F32_16X16X4_F32, reusing the A fragment across both tiles.
// ---------------------------------------------------------------------------
__global__ __launch_bounds__(256)
void wmma_gemm_lds(const float* __restrict__ A, const float* __restrict__ Bm,
                   const float* __restrict__ bias, float* __restrict__ C)
{
  __shared__ __align__(16) float sA[16 * 64];
  __shared__ __align__(16) float sB[64 * 256];

  const int t    = threadIdx.x;
  const int wave = t >> 5;
  const int lane = t & 31;
  const int hi   = lane >> 4;        // half-wave
  const int mrow = lane & 15;
  const int ncol = lane & 15;
  const int m0   = blockIdx.x * 16;
  const int nA   = wave * 16 + ncol; // this lane's column for tile 0 (tile 1: +128)

  // ISA 10.2: generic LDS address low 32 bits == LDS byte offset
  const uint32_t sA_base = (uint32_t)(uintptr_t)sA;
  const uint32_t sB_base = (uint32_t)(uintptr_t)sB;

  v8f acc0 = {0.f,0.f,0.f,0.f,0.f,0.f,0.f,0.f};
  v8f acc1 = {0.f,0.f,0.f,0.f,0.f,0.f,0.f,0.f};

  const int arow = t >> 4, aseg = t & 15;    // A-copy assignment (1 x b128/thread)

  for (int kc = 0; kc < 256; kc += 64) {
    // ---- async stage: B chunk 64x256 (contiguous 64KB), 16 x b128 per thread
    const float* gB = Bm + kc * 256;
    #pragma unroll
    for (int i = 0; i < 16; ++i) {
      const int seg = t + i * 256;           // 16-byte segment index
      async_b128(sB_base + (uint32_t)seg * 16u, gB + seg * 4);
    }
    // ---- async stage: A chunk 16x64 (row stride 256), 1 x b128 per thread
    async_b128(sA_base + (uint32_t)(arow * 64 + aseg * 4) * 4u,
               A + (long)(m0 + arow) * 256 + kc + aseg * 4);

    wait_async0();
    __syncthreads();

    #pragma unroll 4
    for (int kl = 0; kl < 64; kl += 4) {
      const int ka = kl + 2 * hi;
      v2f a;  a.x  = sA[mrow * 64 + ka];        a.y  = sA[mrow * 64 + ka + 1];
      v2f b0; b0.x = sB[ka * 256 + nA];         b0.y = sB[(ka + 1) * 256 + nA];
      v2f b1; b1.x = sB[ka * 256 + nA + 128];   b1.y = sB[(ka + 1) * 256 + nA + 128];
      acc0 = __builtin_amdgcn_wmma_f32_16x16x4_f32(
          false, a, false, b0, (short)0, acc0, false, false);
      acc1 = __builtin_amdgcn_wmma_f32_16x16x4_f32(
          false, a, false, b1, (short)0, acc1, false, false);
    }
    __syncthreads();
  }

  const int mb = m0 + 8 * hi;                 // C/D layout: VGPR r -> row mb+r
  const float bv0 = bias ? bias[nA]       : 0.f;
  const float bv1 = bias ? bias[nA + 128] : 0.f;
  #pragma unroll
  for (int r = 0; r < 8; ++r) {
    C[(long)(mb + r) * 256 + nA]       = acc0[r] + bv0;
    C[(long)(mb + r) * 256 + nA + 128] = acc1[r] + bv1;
  }
}

// ---------------------------------------------------------------------------
// Direct-load WMMA GEMM for small / irregular shapes.  One wave per 16x16
// tile.  A = [A0 | A1] concat along K at K0 (A1==nullptr -> K0 must equal K).
// Out-of-range N columns are computed on clamped (in-bounds) B columns and
// simply not stored (each lane's accumulator holds exactly one column).
// ---------------------------------------------------------------------------
__global__ __launch_bounds__(32)
void wmma_gemm(const float* __restrict__ A0, int a0s,
               const float* __restrict__ A1, int a1s, int K0,
               const float* __restrict__ Bm, int ldb,
               const float* __restrict__ bias,
               float* __restrict__ C, int ldc,
               int N, int K)
{
  const int lane = threadIdx.x & 31;
  const int hi   = lane >> 4;
  const int mrow = lane & 15;
  const int ncol = lane & 15;
  const int m0 = blockIdx.y * 16;
  const int n0 = blockIdx.x * 16;

  const int nb  = n0 + ncol;
  const int nbc = (nb < N) ? nb : (N - 1);   // clamped column (stays in-bounds)

  v8f acc = {0.f,0.f,0.f,0.f,0.f,0.f,0.f,0.f};

  const float* Ar0 = A0 + (long)(m0 + mrow) * a0s;
  #pragma unroll 4
  for (int k0 = 0; k0 < K0; k0 += 4) {
    const int ka = k0 + 2 * hi;
    v2f a;  a.x = Ar0[ka];  a.y = Ar0[ka + 1];
    v2f b;
    b.x = Bm[(long)ka       * ldb + nbc];
    b.y = Bm[(long)(ka + 1) * ldb + nbc];
    acc = __builtin_amdgcn_wmma_f32_16x16x4_f32(
        false, a, false, b, (short)0, acc, false, false);
  }
  if (A1) {                                   // uniform branch: EXEC stays full
    const float* Ar1 = A1 + (long)(m0 + mrow) * a1s - K0;
    #pragma unroll 4
    for (int k0 = K0; k0 < K; k0 += 4) {
      const int ka = k0 + 2 * hi;
      v2f a;  a.x = Ar1[ka];  a.y = Ar1[ka + 1];
      v2f b;
      b.x = Bm[(long)ka       * ldb + nbc];
      b.y = Bm[(long)(ka + 1) * ldb + nbc];
      acc = __builtin_amdgcn_wmma_f32_16x16x4_f32(
          false, a, false, b, (short)0, acc, false, false);
    }
  }

  if (nb < N) {
    const float bv = bias ? bias[nb] : 0.0f;
    const int mb = m0 + 8 * hi;
    #pragma unroll
    for (int r = 0; r < 8; ++r)
      C[(long)(mb + r) * ldc + nb] = acc[r] + bv;
  }
}

// t[b,h] = sum_s tanh(c_slot[b,s,h] + u[b,h])
__global__ __launch_bounds__(256)
void tanhsum_kernel(const float* __restrict__ c_slot,
                    const float* __restrict__ u, float* __restrict__ tsum)
{
  const int idx = blockIdx.x * 256 + threadIdx.x;   // b*256 + h
  const int b = idx >> 8, hh = idx & 255;
  const float uv = u[idx];
  const float* cs = c_slot + (long)b * 128 * 256 + hh;
  float acc = 0.f;
  #pragma unroll 4
  for (int s = 0; s < 128; ++s) acc += fast_tanh(cs[s * 256] + uv);
  tsum[idx] = acc;
}

// r_slot[b,s,h] = g[b,h] * c_slot[b,s,h]
__global__ __launch_bounds__(256)
void rslot_kernel(const float* __restrict__ c_slot,
                  const float* __restrict__ g, float* __restrict__ r_slot)
{
  const long idx = (long)blockIdx.x * 256 + threadIdx.x;
  const int hh = (int)(idx & 255);
  const int b  = (int)(idx >> 15);
  r_slot[idx] = c_slot[idx] * g[b * 256 + hh];
}

// One block per (b,i): scores[b,i,j] = sum_h tanh(sf[b,i,h]+hf[b,j,h])*w[h];
// alpha[b,i] = softmax_j(scores)[i].
__global__ __launch_bounds__(256)
void scores_alpha_kernel(const float* __restrict__ sf,
                         const float* __restrict__ hf,
                         const float* __restrict__ w_ID,
                         float* __restrict__ alphas)
{
  __shared__ float s_sf[256];
  __shared__ float s_w[256];
  __shared__ float s_scores[128];

  const int bi = blockIdx.x;            // b*128 + i
  const int b  = bi >> 7;
  const int i  = bi & 127;
  const int t  = threadIdx.x;
  s_sf[t] = sf[(long)bi * 256 + t];
  s_w[t]  = w_ID[t];
  __syncthreads();

  const int wave = t >> 5, lane = t & 31;
  const float* hfb = hf + (long)b * 128 * 256;
  for (int j = wave; j < 128; j += 8) {
    const float* hfr = hfb + j * 256;
    float p = 0.f;
    #pragma unroll
    for (int hh = lane; hh < 256; hh += 32)
      p += fast_tanh(s_sf[hh] + hfr[hh]) * s_w[hh];
    #pragma unroll
    for (int off = 16; off; off >>= 1) p += __shfl_xor(p, off, 32);
    if (lane == 0) s_scores[j] = p;
  }
  __syncthreads();

  if (wave == 0) {
    float m = -1e30f;
    for (int j = lane; j < 128; j += 32) m = fmaxf(m, s_scores[j]);
    #pragma unroll
    for (int off = 16; off; off >>= 1) m = fmaxf(m, __shfl_xor(m, off, 32));
    float ssum = 0.f;
    for (int j = lane; j < 128; j += 32)
      ssum += __builtin_amdgcn_exp2f((s_scores[j] - m) * 1.4426950408889634f);
    #pragma unroll
    for (int off = 16; off; off >>= 1) ssum += __shfl_xor(ssum, off, 32);
    if (lane == 0) {
      float num = __builtin_amdgcn_exp2f((s_scores[i] - m) * 1.4426950408889634f);
      alphas[bi] = num * __builtin_amdgcn_rcpf(ssum);
    }
  }
}

// r_inte[b,h] = c_inte[b,h] + sum_s r_slot[b,s,h]*alphas[b,s]
__global__ __launch_bounds__(256)
void reduce_r_kernel(const float* __restrict__ r_slot,
                     const float* __restrict__ alphas,
                     const float* __restrict__ c_inte,
                     float* __restrict__ r_inte)
{
  const int idx = blockIdx.x * 256 + threadIdx.x;  // b*256 + h
  const int b = idx >> 8, hh = idx & 255;
  const float* rs = r_slot + (long)b * 128 * 256 + hh;
  const float* al = alphas + b * 128;
  float acc = 0.f;
  #pragma unroll 4
  for (int s = 0; s < 128; ++s) acc += rs[s * 256] * al[s];
  r_inte[idx] = acc + c_inte[idx];
}

extern "C" void kernel_launch(void* const* d_in, const int* in_sizes, int n_in,
                              void* d_out, int out_size, void* d_ws, size_t ws_size,
                              hipStream_t stream)
{
  (void)in_sizes; (void)n_in; (void)out_size; (void)ws_size;
  const float* h       = (const float*)d_in[0];
  const float* c_slot  = (const float*)d_in[1];
  const float* c_inte  = (const float*)d_in[2];
  const float* W_SF    = (const float*)d_in[3];
  const float* V_SF    = (const float*)d_in[4];
  const float* V1_ID   = (const float*)d_in[5];
  const float* V2_ID_w = (const float*)d_in[6];
  const float* V2_ID_b = (const float*)d_in[7];
  const float* w_ID    = (const float*)d_in[8];
  const float* W_inte  = (const float*)d_in[9];
  const float* W_slot  = (const float*)d_in[10];

  constexpr int B = 16, S = 128, H = 256, NL = 22, NS = 122;

  float* ws     = (float*)d_ws;
  float* hf     = ws;  ws += B * S * H;
  float* r_slot = ws;  ws += B * S * H;
  float* sf     = ws;  ws += B * S * H;
  float* r_inte = ws;  ws += B * H;
  float* u      = ws;  ws += B * H;
  float* tsum   = ws;  ws += B * H;
  float* g      = ws;  ws += B * H;
  float* alphas = ws;  ws += B * S;

  // hf = h @ V2_ID_w + V2_ID_b   (loop invariant) — async-LDS-staged WMMA
  wmma_gemm_lds<<<(B * S) / 16, 256, 0, stream>>>(h, V2_ID_w, V2_ID_b, hf);

  hipMemcpyAsync(r_inte, c_inte, (size_t)B * H * sizeof(float),
                 hipMemcpyDeviceToDevice, stream);

  for (int it = 0; it < 3; ++it) {
    // u = r_inte @ W_SF
    wmma_gemm<<<dim3(H / 16, B / 16), 32, 0, stream>>>(
        r_inte, H, nullptr, 0, H, W_SF, H, nullptr, u, H, H, H);
    // tsum = sum_s tanh(c_slot + u)
    tanhsum_kernel<<<B * H / 256, 256, 0, stream>>>(c_slot, u, tsum);
    // g = tsum @ V_SF     ( == (tanh(...)@V_SF).sum(1) )
    wmma_gemm<<<dim3(H / 16, B / 16), 32, 0, stream>>>(
        tsum, H, nullptr, 0, H, V_SF, H, nullptr, g, H, H, H);
    // r_slot = g * c_slot
    rslot_kernel<<<B * S * H / 256, 256, 0, stream>>>(c_slot, g, r_slot);
    // sf = r_slot @ V1_ID — async-LDS-staged WMMA
    wmma_gemm_lds<<<(B * S) / 16, 256, 0, stream>>>(r_slot, V1_ID, nullptr, sf);
    // alphas = diag of row-softmax(scores)
    scores_alpha_kernel<<<B * S, 256, 0, stream>>>(sf, hf, w_ID, alphas);
    // r_inte = c_inte + sum_s r_slot*alphas
    reduce_r_kernel<<<B * H / 256, 256, 0, stream>>>(r_slot, alphas, c_inte, r_inte);
  }

  float* slot_out = (float*)d_out;            // (B,S,NS)
  float* inte_out = slot_out + B * S * NS;    // (B,NL)

  // slot_output = concat([h, r_slot], -1) @ W_slot_ans
  wmma_gemm<<<dim3((NS + 15) / 16, (B * S) / 16), 32, 0, stream>>>(
      h, H, r_slot, H, H, W_slot, NS, nullptr, slot_out, NS, NS, 2 * H);
  // intent_output = concat([r_inte, h[:,S-1,:]], -1) @ W_inte_ans
  wmma_gemm<<<dim3((NL + 15) / 16, B / 16), 32, 0, stream>>>(
      r_inte, H, h + (long)(S - 1) * H, S * H, H, W_inte, NL, nullptr,
      inte_out, NL, NL, 2 * H);
}